// FixedQCDAware_68813966017009
// MI455X (gfx1250) — compile-verified
//
#include <hip/hip_runtime.h>

typedef float v2f __attribute__((ext_vector_type(2)));
typedef float v8f __attribute__((ext_vector_type(8)));

#define B_      8
#define F_      64
#define N_      2048
#define EPSILON 1e-7f

// ---------------------------------------------------------------------------
// Kernel 1: per-batch column stats.
//   norms[b,i] = sum_f E[f,i]^2
//   pows [b,i] = E[4,i]^2            (pow_momenta, ALPHA=1)
//   dmin [b]   = min_i pows[b,i] + EPSILON
// One block per batch, 1024 threads, LDS tree-reduce for the min.
// ---------------------------------------------------------------------------
__global__ void qcd_stats_kernel(const float* __restrict__ emb,
                                 float* __restrict__ norms,
                                 float* __restrict__ pows,
                                 float* __restrict__ dmin) {
    const int b = blockIdx.x;
    const float* E = emb + (size_t)b * F_ * N_;
    const int t = threadIdx.x;

    float lmin = 3.402823466e38f;
    for (int c = t; c < N_; c += 1024) {
        float s = 0.0f;
        #pragma unroll 8
        for (int f = 0; f < F_; ++f) {
            float v = E[f * N_ + c];
            s = fmaf(v, v, s);
        }
        float p = E[4 * N_ + c];
        p = p * p;
        norms[b * N_ + c] = s;
        pows[b * N_ + c] = p;
        lmin = fminf(lmin, p);
    }

    __shared__ float sm[1024];
    sm[t] = lmin;
    __syncthreads();
    for (int s = 512; s > 0; s >>= 1) {
        if (t < s) sm[t] = fminf(sm[t], sm[t + s]);
        __syncthreads();
    }
    if (t == 0) dmin[b] = sm[0] + EPSILON;
}

// ---------------------------------------------------------------------------
// Kernel 2: fused Gram (WMMA f32 16x16x4) + QCD-aware epilogue.
// One wave32 computes a 32x32 tile of w_ij:
//   G[i,j]   = sum_f E[f,i] * E[f,j]         (4 accumulators, 16 K-steps)
//   sq       = norms[i] + norms[j] - 2*G
//   w        = exp(-(sq*min(p_i,p_j) - dmin)/dmin)
//
// A-matrix (16x4 f32) lane layout: lane L(<16) row M=L, VGPR0=K0/VGPR1=K1;
// lanes 16-31 hold K2/K3.  B (4x16) mirrors with N striped across lanes.
// C/D (16x16 f32): VGPR r -> row r + 8*(lane>=16), col = lane&15.
// ---------------------------------------------------------------------------
__global__ void __launch_bounds__(128)
qcd_gram_wmma_kernel(const float* __restrict__ emb,
                     const float* __restrict__ norms,
                     const float* __restrict__ pows,
                     const float* __restrict__ dmin,
                     float* __restrict__ out) {
    const int b    = blockIdx.y;
    const int wave = threadIdx.x >> 5;
    const int lane = threadIdx.x & 31;
    const int tile = blockIdx.x * 4 + wave;      // 0..4095
    const int ti   = tile >> 6;                  // 0..63  (row block of 32)
    const int tj   = tile & 63;                  // 0..63  (col block of 32)
    const int I    = ti * 32;
    const int J    = tj * 32;

    const float* E = emb + (size_t)b * F_ * N_;
    const int half = lane >> 4;                  // 0: K={0,1}, 1: K={2,3}
    const int lc   = lane & 15;

    v8f acc00 = {}, acc01 = {}, acc10 = {}, acc11 = {};

    #pragma unroll 4
    for (int kk = 0; kk < 16; ++kk) {
        const float* r0 = E + (size_t)(4 * kk + 2 * half) * N_;
        const float* r1 = r0 + N_;

        if (kk < 15) {
            __builtin_prefetch(r0 + 4 * N_ + I + lc, 0, 0);
            __builtin_prefetch(r0 + 4 * N_ + J + lc, 0, 0);
        }

        v2f aI0 = { r0[I + lc],      r1[I + lc]      };
        v2f aI1 = { r0[I + 16 + lc], r1[I + 16 + lc] };
        v2f bJ0 = { r0[J + lc],      r1[J + lc]      };
        v2f bJ1 = { r0[J + 16 + lc], r1[J + 16 + lc] };

        acc00 = __builtin_amdgcn_wmma_f32_16x16x4_f32(false, aI0, false, bJ0,
                                                      (short)0, acc00, false, false);
        acc01 = __builtin_amdgcn_wmma_f32_16x16x4_f32(false, aI0, false, bJ1,
                                                      (short)0, acc01, false, false);
        acc10 = __builtin_amdgcn_wmma_f32_16x16x4_f32(false, aI1, false, bJ0,
                                                      (short)0, acc10, false, false);
        acc11 = __builtin_amdgcn_wmma_f32_16x16x4_f32(false, aI1, false, bJ1,
                                                      (short)0, acc11, false, false);
    }

    const float dm = dmin[b];
    const float inv_dm = 1.0f / dm;
    const float* nb = norms + b * N_;
    const float* pb = pows + b * N_;
    float* ob = out + (size_t)b * N_ * N_;

    // column-side (N) quantities: col = base + lc for all 8 VGPRs of an acc
    const float ncJ0 = nb[J + lc],      pcJ0 = pb[J + lc];
    const float ncJ1 = nb[J + 16 + lc], pcJ1 = pb[J + 16 + lc];

    #pragma unroll
    for (int r = 0; r < 8; ++r) {
        const int row0 = I + r + 8 * half;          // acc00 / acc01 row
        const int row1 = row0 + 16;                 // acc10 / acc11 row
        const float nr0 = nb[row0], pr0 = pb[row0];
        const float nr1 = nb[row1], pr1 = pb[row1];

        {   // acc00: rows I.., cols J..
            float sq = nr0 + ncJ0 - 2.0f * acc00[r];
            float w  = __expf(-(sq * fminf(pr0, pcJ0) - dm) * inv_dm);
            ob[(size_t)row0 * N_ + J + lc] = w;
        }
        {   // acc01: rows I.., cols J+16..
            float sq = nr0 + ncJ1 - 2.0f * acc01[r];
            float w  = __expf(-(sq * fminf(pr0, pcJ1) - dm) * inv_dm);
            ob[(size_t)row0 * N_ + J + 16 + lc] = w;
        }
        {   // acc10: rows I+16.., cols J..
            float sq = nr1 + ncJ0 - 2.0f * acc10[r];
            float w  = __expf(-(sq * fminf(pr1, pcJ0) - dm) * inv_dm);
            ob[(size_t)row1 * N_ + J + lc] = w;
        }
        {   // acc11: rows I+16.., cols J+16..
            float sq = nr1 + ncJ1 - 2.0f * acc11[r];
            float w  = __expf(-(sq * fminf(pr1, pcJ1) - dm) * inv_dm);
            ob[(size_t)row1 * N_ + J + 16 + lc] = w;
        }
    }
}

// ---------------------------------------------------------------------------
extern "C" void kernel_launch(void* const* d_in, const int* in_sizes, int n_in,
                              void* d_out, int out_size, void* d_ws, size_t ws_size,
                              hipStream_t stream) {
    (void)in_sizes; (void)n_in; (void)out_size; (void)ws_size;

    const float* emb = (const float*)d_in[0];
    float* out = (float*)d_out;

    float* norms = (float*)d_ws;               // 8 * 2048
    float* pows  = norms + B_ * N_;            // 8 * 2048
    float* dmin  = pows + B_ * N_;             // 8

    qcd_stats_kernel<<<B_, 1024, 0, stream>>>(emb, norms, pows, dmin);

    dim3 grid(1024, B_);   // 1024 blocks * 4 waves = 4096 tiles of 32x32 per batch
    qcd_gram_wmma_kernel<<<grid, 128, 0, stream>>>(emb, norms, pows, dmin, out);
}